// NBFNet_33689723470229
// MI455X (gfx1250) — compile-verified
//
#include <hip/hip_runtime.h>
#include <hip/hip_bf16.h>

#define DD 32
#define KDIM (13*DD)        // 416
#define LDS_STRIDE 420      // 420*4B: rows land on distinct banks (gcd(36,64)=4, 16 distinct)
#define NNODES 20000        // num_nodes is a device scalar; grid dims must be host-known

typedef __attribute__((ext_vector_type(2))) float v2f;
typedef __attribute__((ext_vector_type(8))) float v8f;

// ---- float atomic max/min via monotonic integer reinterpretation (order-safe, no CAS loop)
__device__ __forceinline__ void atomicMaxF(float* addr, float v) {
  if (v >= 0.0f) atomicMax((int*)addr, __float_as_int(v));
  else           atomicMin((unsigned int*)addr, __float_as_uint(v));
}
__device__ __forceinline__ void atomicMinF(float* addr, float v) {
  if (v >= 0.0f) atomicMin((int*)addr, __float_as_int(v));
  else           atomicMax((unsigned int*)addr, __float_as_uint(v));
}

// ---- setup: deg=1, gather query rows, zero log-sum accumulator
__global__ void k_setup(const int* r_index, const float* query_emb,
                        float* q, float* deg, float* dsum, int N, int B) {
  int tid = blockIdx.x * blockDim.x + threadIdx.x;
  if (tid < N) deg[tid] = 1.0f;
  if (tid < B * DD) {
    int b = tid / DD, d = tid % DD;
    q[tid] = query_emb[r_index[b] * DD + d];
  }
  if (tid == 0) *dsum = 0.0f;
}

__global__ void k_deg(const int* ei, float* deg, int E) {
  int e = blockIdx.x * blockDim.x + threadIdx.x;
  if (e < E) atomicAdd(&deg[ei[E + e]], 1.0f);
}

__global__ void k_dlogsum(const float* deg, float* dsum, int N) {
  __shared__ float red[256];
  int tid = blockIdx.x * 256 + threadIdx.x;
  red[threadIdx.x] = (tid < N) ? __logf(deg[tid]) : 0.0f;
  __syncthreads();
  for (int st = 128; st > 0; st >>= 1) {
    if ((int)threadIdx.x < st) red[threadIdx.x] += red[threadIdx.x + st];
    __syncthreads();
  }
  if (threadIdx.x == 0) atomicAdd(dsum, red[0]);
}

__global__ void k_scale(const float* deg, const float* dsum, float* scl, int N) {
  int n = blockIdx.x * blockDim.x + threadIdx.x;
  if (n < N) scl[n] = __logf(deg[n]) * (float)N / *dsum;
}

// ---- rel[b,t,d] = query[b,:] . rel_W[t*D+d,:] + rel_b[t*D+d]
__global__ void k_rel(const float* q, const float* W, const float* bvec, float* rel,
                      int B, int R) {
  int tid = blockIdx.x * blockDim.x + threadIdx.x;
  if (tid >= B * R * DD) return;
  int b = tid / (R * DD);
  int row = tid - b * (R * DD);
  float acc = bvec[row];
  const float* wr = W + (size_t)row * DD;
  const float* qb = q + b * DD;
#pragma unroll
  for (int j = 0; j < DD; j++) acc += qb[j] * wr[j];
  rel[tid] = acc;
}

// ---- init aggregation buffers to the boundary message (it is one of the segment members)
__global__ void k_agg_init(const int* h_index, const float* q,
                           float* x, float* s, float* sq, float* mx, float* mn,
                           int N, int B) {
  int tid = blockIdx.x * blockDim.x + threadIdx.x;
  if (tid >= B * N * DD) return;
  int b = tid / (N * DD);
  int rem = tid - b * (N * DD);
  int n = rem / DD;
  int d = rem - n * DD;
  float v = (n == h_index[b]) ? q[b * DD + d] : 0.0f;
  if (x) x[tid] = v;
  s[tid] = v; sq[tid] = v * v; mx[tid] = v; mn[tid] = v;
}

// ---- edge messages + PNA scatter: one 32-lane vector per edge, loop over batch
__global__ void k_edges(const int* ei, const int* et, const float* x, const float* rel,
                        float* s, float* sq, float* mx, float* mn,
                        int E, int N, int B, int R) {
  int tid = blockIdx.x * blockDim.x + threadIdx.x;
  int e = tid >> 5;
  int d = tid & 31;
  if (e >= E) return;
  int sn = ei[e];
  int dn = ei[E + e];
  int t  = et[e];
  for (int b = 0; b < B; b++) {
    size_t si = ((size_t)b * N + sn) * DD + d;
    size_t di = ((size_t)b * N + dn) * DD + d;
    float m = x[si] * rel[(size_t)(b * R + t) * DD + d];
    atomicAdd(&s[di], m);
    atomicAdd(&sq[di], m * m);
    atomicMaxF(&mx[di], m);
    atomicMinF(&mn[di], m);
  }
}

// ---- per-node PNA features + conv GEMM via fp32 WMMA (16x16x4), bias + ReLU
// One wave per 16-row tile; 13D=416 input staged in LDS; both 16-col output tiles.
__global__ __launch_bounds__(32) void k_conv(
    const float* x, const float* s, const float* sq, const float* mx, const float* mn,
    const float* deg, const float* scl, const float* W, const float* bias,
    float* out, int N, int BN) {
  __shared__ float lds[16 * LDS_STRIDE];
  const int lane = threadIdx.x;
  const int row0 = blockIdx.x * 16;
  const int d = lane;           // lane owns feature dim d for all 16 rows

  for (int r = 0; r < 16; r++) {
    int gr = row0 + r;
    float xv = 0.f, f0 = 0.f, f1 = 0.f, f2 = 0.f, f3 = 0.f, sc1 = 1.f, sc2 = 1.f;
    if (gr < BN) {
      int n = gr % N;
      size_t idx = (size_t)gr * DD + d;
      xv = x[idx];
      float dg   = deg[n];
      float mean = s[idx] / dg;
      float sqm  = sq[idx] / dg;
      f0 = mean;
      f1 = mx[idx];
      f2 = mn[idx];
      f3 = __fsqrt_rn(fmaxf(sqm - mean * mean, 1e-6f));
      float scn = scl[n];
      sc1 = scn;
      sc2 = 1.0f / fmaxf(scn, 0.01f);
    }
    lds[r * LDS_STRIDE + d] = xv;
    float feats[4] = {f0, f1, f2, f3};
    float scs[3]   = {1.0f, sc1, sc2};
#pragma unroll
    for (int st = 0; st < 4; st++)
#pragma unroll
      for (int c = 0; c < 3; c++)
        lds[r * LDS_STRIDE + DD + (d * 4 + st) * 3 + c] = feats[st] * scs[c];
  }
  __syncthreads();

  const int half = lane >> 4;       // K split half per ISA A/B layout
  const int l16  = lane & 15;
  const float* Wrow0 = W + (size_t)l16 * KDIM;          // B[k][j] = W[j][k]
  const float* Wrow1 = W + (size_t)(l16 + 16) * KDIM;
  const float* arow  = lds + l16 * LDS_STRIDE;
  __builtin_prefetch(Wrow0, 0, 1);   // global_prefetch_b8
  __builtin_prefetch(Wrow1, 0, 1);

  v8f acc0 = {};
  v8f acc1 = {};
  for (int k = 0; k < KDIM; k += 4) {
    int kk = k + 2 * half;
    v2f a;  a.x  = arow[kk];   a.y  = arow[kk + 1];
    v2f b0; b0.x = Wrow0[kk];  b0.y = Wrow0[kk + 1];
    v2f b1; b1.x = Wrow1[kk];  b1.y = Wrow1[kk + 1];
    acc0 = __builtin_amdgcn_wmma_f32_16x16x4_f32(false, a, false, b0, (short)0, acc0, false, false);
    acc1 = __builtin_amdgcn_wmma_f32_16x16x4_f32(false, a, false, b1, (short)0, acc1, false, false);
  }

  // C/D layout: VGPR v, lanes 0-15 -> M=v, lanes 16-31 -> M=v+8; N=lane&15
  float bz0 = bias[l16], bz1 = bias[l16 + 16];
#pragma unroll
  for (int v = 0; v < 8; v++) {
    int gr = row0 + v + 8 * half;
    if (gr < BN) {
      out[(size_t)gr * DD + l16]      = fmaxf(acc0[v] + bz0, 0.0f);
      out[(size_t)gr * DD + l16 + 16] = fmaxf(acc1[v] + bz1, 0.0f);
    }
  }
}

// ---- final MLP on gathered tails: feature=[x(D) | query(D)] -> relu(64x64) -> dot(64)
__global__ __launch_bounds__(64) void k_mlp(
    const float* x, const float* q, const int* t_index,
    const float* W1, const float* b1, const float* W2, const float* b2,
    float* out, int N, int K) {
  __shared__ float f[2 * DD];
  __shared__ float red[2 * DD];
  int bk = blockIdx.x;
  int b = bk / K, k = bk % K;
  int j = threadIdx.x;
  int node = t_index[b * K + k];
  f[j] = (j < DD) ? x[((size_t)b * N + node) * DD + j] : q[b * DD + (j - DD)];
  __syncthreads();
  float acc = b1[j];
#pragma unroll 8
  for (int i = 0; i < 2 * DD; i++) acc += f[i] * W1[j * (2 * DD) + i];
  red[j] = fmaxf(acc, 0.0f) * W2[j];
  __syncthreads();
  for (int st = DD; st > 0; st >>= 1) {
    if (j < st) red[j] += red[j + st];
    __syncthreads();
  }
  if (j == 0) out[bk] = red[0] + b2[0];
}

extern "C" void kernel_launch(void* const* d_in, const int* in_sizes, int n_in,
                              void* d_out, int out_size, void* d_ws, size_t ws_size,
                              hipStream_t stream) {
  const int* edge_index = (const int*)d_in[0];
  const int* edge_type  = (const int*)d_in[1];
  const int* h_index    = (const int*)d_in[2];
  const int* t_index    = (const int*)d_in[3];
  const int* r_index    = (const int*)d_in[4];
  const float* query_emb = (const float*)d_in[6];
  const float* rel_W  = (const float*)d_in[7];
  const float* rel_b  = (const float*)d_in[8];
  const float* conv_W = (const float*)d_in[9];
  const float* conv_b = (const float*)d_in[10];
  const float* mlp_W1 = (const float*)d_in[11];
  const float* mlp_b1 = (const float*)d_in[12];
  const float* mlp_W2 = (const float*)d_in[13];
  const float* mlp_b2 = (const float*)d_in[14];

  const int N = NNODES;
  const int E = in_sizes[1];
  const int B = in_sizes[2];
  const int K = in_sizes[3] / B;
  const int R = in_sizes[6] / DD;
  const int L = in_sizes[8] / (R * DD);

  // scratch layout (256B aligned)
  size_t off = 0;
  auto alloc = [&](size_t cnt) -> float* {
    float* p = (float*)((char*)d_ws + off);
    off += ((cnt * sizeof(float) + 255) / 256) * 256;
    return p;
  };
  float* q    = alloc((size_t)B * DD);
  float* deg  = alloc((size_t)N);
  float* dsum = alloc(1);
  float* scl  = alloc((size_t)N);
  float* rel  = alloc((size_t)B * R * DD);
  float* nb[5];
  for (int i = 0; i < 5; i++) nb[i] = alloc((size_t)B * N * DD);
  (void)ws_size; (void)n_in; (void)out_size;

  const int thr = 256;
  const int gN = (N + thr - 1) / thr;
  k_setup<<<gN, thr, 0, stream>>>(r_index, query_emb, q, deg, dsum, N, B);
  k_deg<<<(E + thr - 1) / thr, thr, 0, stream>>>(edge_index, deg, E);
  k_dlogsum<<<gN, thr, 0, stream>>>(deg, dsum, N);
  k_scale<<<gN, thr, 0, stream>>>(deg, dsum, scl, N);

  float* x  = nb[0];
  float* s  = nb[1];
  float* sq = nb[2];
  float* mx = nb[3];
  float* mn = nb[4];
  const int gBND = (B * N * DD + thr - 1) / thr;
  const int gE32 = (E * 32 + thr - 1) / thr;
  const int tiles = (B * N + 15) / 16;
  for (int l = 0; l < L; l++) {
    k_rel<<<(B * R * DD + thr - 1) / thr, thr, 0, stream>>>(
        q, rel_W + (size_t)l * R * DD * DD, rel_b + (size_t)l * R * DD, rel, B, R);
    k_agg_init<<<gBND, thr, 0, stream>>>(h_index, q, (l == 0) ? x : (float*)nullptr,
                                         s, sq, mx, mn, N, B);
    k_edges<<<gE32, thr, 0, stream>>>(edge_index, edge_type, x, rel, s, sq, mx, mn,
                                      E, N, B, R);
    // conv output aliases s: each wave reads its 16 rows before writing them
    k_conv<<<tiles, 32, 0, stream>>>(x, s, sq, mx, mn, deg, scl,
                                     conv_W + (size_t)l * DD * KDIM,
                                     conv_b + (size_t)l * DD, s, N, B * N);
    float* t = x; x = s; s = t;  // new x = conv output; old x recycled as next s
  }
  k_mlp<<<B * K, 64, 0, stream>>>(x, q, t_index, mlp_W1, mlp_b1, mlp_W2, mlp_b2,
                                  (float*)d_out, N, K);
}